// ForwardBPTTCell_70566312673291
// MI455X (gfx1250) — compile-verified
//
#include <hip/hip_runtime.h>
#include <math.h>

// ForwardBPTT GRU cell for MI455X (gfx1250, wave32, WMMA).
// T=512 sequential steps; heavy parts: prod_jac chain (f32 WMMA 16x16x4 with
// LDS-staged, software-pipelined J panels), exact per-step 256x256 solve
// (LDS Gaussian elimination), parallel jacT materialization (bandwidth-bound,
// then L2-resident: 136MB < 192MB global L2).

#define TT     512
#define DIN    128
#define HH     256
#define H3     768
#define DOUT   128
#define LN_EPS 1e-6f

typedef __attribute__((ext_vector_type(2))) float v2f;
typedef __attribute__((ext_vector_type(8))) float v8f;

// ---------------------------------------------------------------- reductions
__device__ __forceinline__ float reduce256(float* arr) {
  const int tid = threadIdx.x;
  for (int s = 128; s > 0; s >>= 1) {
    __syncthreads();
    if (tid < s) arr[tid] += arr[tid + s];
  }
  __syncthreads();
  float v = arr[0];
  __syncthreads();
  return v;
}

// ------------------------------------------------- kernel 0: gx = x@Wx + b
__global__ void __launch_bounds__(256)
k_gx(const float* __restrict__ x, const float* __restrict__ Wx,
     const float* __restrict__ b, float* __restrict__ gx) {
  int idx = blockIdx.x * blockDim.x + threadIdx.x;
  if (idx >= TT * H3) return;
  int t = idx / H3, k = idx % H3;
  const float* xr = x + t * DIN;
  float s = b[k];
#pragma unroll 4
  for (int j = 0; j < DIN; ++j) s = fmaf(xr[j], Wx[j * H3 + k], s);
  gx[idx] = s;
}

// ------------------------- kernel 1: sequential recurrence + readout + VJP
__global__ void __launch_bounds__(768)
k_recur(const float* __restrict__ dout_seq, const float* __restrict__ Wh,
        const float* __restrict__ lns, const float* __restrict__ lnb,
        const float* __restrict__ Wo, const float* __restrict__ bo,
        const float* __restrict__ gx_all,
        float* __restrict__ o_out, float* __restrict__ o_prevh,
        float* __restrict__ o_inst, float* __restrict__ o_hnorm,
        float* __restrict__ cA, float* __restrict__ cB,
        float* __restrict__ cC, float* __restrict__ cZ) {
  __shared__ float h[HH], nh[HH], gh[H3], xh[HH], ra[HH], rb[HH];
  const int tid = threadIdx.x;
  if (tid < HH) h[tid] = 0.f;
  __syncthreads();

  for (int t = 0; t < TT; ++t) {
    {  // gh = h @ Wh
      float s = 0.f;
      const float* w = Wh + tid;
      for (int j = 0; j < HH; ++j) s = fmaf(h[j], w[j * H3], s);
      gh[tid] = s;
    }
    __syncthreads();

    if (tid < HH) {
      const float* gxr = gx_all + (size_t)t * H3;
      float gz = gxr[tid] + gh[tid];
      float gr = gxr[HH + tid] + gh[HH + tid];
      float hn = gh[2 * HH + tid];
      float z = 1.f / (1.f + expf(-gz));
      float r = 1.f / (1.f + expf(-gr));
      float n = tanhf(gxr[2 * HH + tid] + r * hn);
      float hp = h[tid];
      float newh = (1.f - z) * n + z * hp;
      // jacT[j,i] = Whz[j,i]*a_i + Whr[j,i]*c_i + Whn[j,i]*b_i + (i==j)*z_i
      float omz = 1.f - z, nn2 = 1.f - n * n;
      size_t ci = (size_t)t * HH + tid;
      cA[ci] = (hp - n) * z * omz;
      cB[ci] = omz * nn2 * r;
      cC[ci] = omz * nn2 * hn * r * (1.f - r);
      cZ[ci] = z;
      o_prevh[ci] = hp;
      nh[tid] = newh;
      ra[tid] = newh;
      rb[tid] = newh * newh;
    }
    float s1 = reduce256(ra);
    float s2 = reduce256(rb);
    if (tid == 0) o_hnorm[t] = sqrtf(s2);
    float mu = s1 / (float)HH;
    float var = s2 / (float)HH - mu * mu;
    float invs = rsqrtf(var + LN_EPS);

    if (tid < HH) {
      float xhat = (nh[tid] - mu) * invs;
      xh[tid] = xhat;
      ra[tid] = xhat * lns[tid] + lnb[tid];
    }
    __syncthreads();
    if (tid < DOUT) {
      float acc = bo[tid];
      for (int j = 0; j < HH; ++j) acc = fmaf(ra[j], Wo[j * DOUT + tid], acc);
      o_out[(size_t)t * DOUT + tid] = acc;
    }
    float dxh = 0.f;
    if (tid < HH) {
      float g = 0.f;
      const float* wr = Wo + tid * DOUT;
      const float* dr = dout_seq + (size_t)t * DOUT;
      for (int o = 0; o < DOUT; ++o) g = fmaf(wr[o], dr[o], g);
      dxh = g * lns[tid];
      rb[tid] = dxh;
    }
    float m1 = reduce256(rb) / (float)HH;
    if (tid < HH) rb[tid] = dxh * xh[tid];
    float m2 = reduce256(rb) / (float)HH;
    if (tid < HH)
      o_inst[(size_t)t * HH + tid] = (dxh - m1 - xh[tid] * m2) * invs;

    __syncthreads();
    if (tid < HH) h[tid] = nh[tid];
    __syncthreads();
  }
}

// --------------------- kernel 2: materialize jacT_t for all t (parallel)
__global__ void __launch_bounds__(256)
k_jac(const float* __restrict__ Wh, const float* __restrict__ cA,
      const float* __restrict__ cB, const float* __restrict__ cC,
      const float* __restrict__ cZ, float* __restrict__ jac) {
  int t = blockIdx.x, i = threadIdx.x;
  size_t ci = (size_t)t * HH + i;
  float a = cA[ci], b = cB[ci], c = cC[ci], z = cZ[ci];
  float* J = jac + (size_t)t * HH * HH;
  for (int j = 0; j < HH; ++j) {
    const float* w = Wh + j * H3;
    float v = fmaf(w[i], a, fmaf(w[HH + i], c, w[2 * HH + i] * b));
    if (i == j) v += z;
    J[(size_t)j * HH + i] = v;
  }
}

// -------- kernel 3: prod_jac chain, P(t) = P(t-1) @ J(t)
// P (256x258) lives in LDS; J streamed in 16-row K-panels, double-buffered
// and TRANSPOSED into LDS so each B fragment is one aligned ds_load_b64.
// Software pipeline: global_load_b128 of panel p+1 overlaps WMMAs of panel p.
#define PSTR 258
#define BT   18   // transposed panel stride: [256 cols][16 k] + pad
__global__ void __launch_bounds__(1024)
k_prodjac(const float* __restrict__ jac, float* __restrict__ o_pjn) {
  extern __shared__ float sm[];
  float* P    = sm;                       // 256*258
  float* bufA = sm + 256 * PSTR;          // 256*BT
  float* bufB = bufA + 256 * BT;          // 256*BT
  float* red  = bufB + 256 * BT;          // 1024
  const int tid = threadIdx.x;
  for (int idx = tid; idx < 256 * PSTR; idx += 1024) P[idx] = 0.f;
  __syncthreads();
  if (tid < 256) P[tid * PSTR + tid] = 1.f;  // P = I
  __syncthreads();

  const int wave = tid >> 5, lane = tid & 31;
  const int mi = wave >> 1;            // output tile row 0..15
  const int nb = (wave & 1) * 8;       // tile-col group base
  const int lm = lane & 15, g = lane >> 4;
  const int srow = tid >> 6;           // staging: k-row 0..15
  const int scol = (tid & 63) * 4;     // staging: col (float4 granule)

  for (int t = 0; t < TT; ++t) {
    const float* J = jac + (size_t)t * (HH * HH);
    // keep next step's 256KB Jacobian L2-hot (global_prefetch_b8)
    if (t + 1 < TT) {
      const char* nj = (const char*)(jac + (size_t)(t + 1) * (HH * HH));
      __builtin_prefetch(nj + (size_t)tid * 256, 0, 1);
    }
    v8f acc[8];
#pragma unroll
    for (int q = 0; q < 8; ++q)
      acc[q] = (v8f){0.f, 0.f, 0.f, 0.f, 0.f, 0.f, 0.f, 0.f};

    // prologue: fetch panel 0
    float4 reg = *(const float4*)&J[(size_t)srow * HH + scol];
    for (int p = 0; p < 16; ++p) {
      float* buf = (p & 1) ? bufB : bufA;
      // deposit staged panel, transposed: buf[col][k]
      buf[(scol + 0) * BT + srow] = reg.x;
      buf[(scol + 1) * BT + srow] = reg.y;
      buf[(scol + 2) * BT + srow] = reg.z;
      buf[(scol + 3) * BT + srow] = reg.w;
      __syncthreads();
      if (p < 15)  // prefetch next panel into registers (overlaps WMMAs)
        reg = *(const float4*)&J[(size_t)(16 * (p + 1) + srow) * HH + scol];
#pragma unroll
      for (int q = 0; q < 4; ++q) {
        int kc = (p * 4 + q) * 4 + 2 * g;   // global K for A fragment
        int kl = q * 4 + 2 * g;             // local K inside panel
        v2f a = *(const v2f*)&P[(mi * 16 + lm) * PSTR + kc];
#pragma unroll
        for (int w2 = 0; w2 < 8; ++w2) {
          int cn = (nb + w2) * 16 + lm;
          v2f b = *(const v2f*)&buf[cn * BT + kl];  // one ds_load_b64
          acc[w2] = __builtin_amdgcn_wmma_f32_16x16x4_f32(
              false, a, false, b, (short)0, acc[w2], false, false);
        }
      }
      __syncthreads();
    }
    // all reads of P complete -> overwrite with P@J, accumulate Frobenius
    float ss = 0.f;
#pragma unroll
    for (int q = 0; q < 8; ++q) {
      int cn = (nb + q) * 16 + lm;
#pragma unroll
      for (int v = 0; v < 8; ++v) {
        float val = acc[q][v];
        P[(mi * 16 + v + 8 * g) * PSTR + cn] = val;
        ss += val * val;
      }
    }
    red[tid] = ss;
    __syncthreads();
    for (int s = 512; s > 0; s >>= 1) {
      if (tid < s) red[tid] += red[tid + s];
      __syncthreads();
    }
    if (tid == 0) o_pjn[t] = sqrtf(red[0]);
    __syncthreads();
  }
}

// --- kernel 4: delta chain, exact solve jacT_t x = (delta - inst_delta)
__global__ void __launch_bounds__(1024)
k_delta(const float* __restrict__ jac, const float* __restrict__ inst_seq,
        float* __restrict__ o_delta, float* __restrict__ o_dn) {
  extern __shared__ float sm[];
  float* A = sm;                      // 256 x 258 (col 256 = rhs)
  float* xv = sm + 256 * PSTR;
  float* dprev = xv + 256;
  float* red = dprev + 256;
  const int tid = threadIdx.x;
  if (tid < HH) dprev[tid] = 0.f;
  __syncthreads();

  for (int t = 0; t < TT; ++t) {
    const float* J = jac + (size_t)t * (HH * HH);
    for (int idx = tid; idx < HH * HH; idx += 1024)
      A[(idx >> 8) * PSTR + (idx & 255)] = J[idx];
    if (tid < HH)
      A[tid * PSTR + 256] =
          dprev[tid] - ((t > 0) ? inst_seq[(size_t)(t - 1) * HH + tid] : 0.f);
    __syncthreads();

    for (int p = 0; p < HH - 1; ++p) {
      float pinv = 1.f / A[p * PSTR + p];
      int sub = tid & 3;
      for (int rr = p + 1 + (tid >> 2); rr < HH; rr += 256) {
        float f = A[rr * PSTR + p] * pinv;
        for (int cc = p + 1 + sub; cc <= 256; cc += 4)
          A[rr * PSTR + cc] = fmaf(-f, A[p * PSTR + cc], A[rr * PSTR + cc]);
      }
      __syncthreads();
    }
    if (tid < 32) {  // back substitution: wave 0, wave32 lockstep
      volatile float* xvv = xv;
      for (int p = HH - 1; p >= 0; --p) {
        float partial = 0.f;
        for (int cc = p + 1 + tid; cc < HH; cc += 32)
          partial = fmaf(A[p * PSTR + cc], xvv[cc], partial);
        for (int off = 16; off > 0; off >>= 1)
          partial += __shfl_down(partial, off, 32);
        if (tid == 0)
          xvv[p] = (A[p * PSTR + 256] - partial) / A[p * PSTR + p];
      }
    }
    __syncthreads();
    float ss = 0.f;
    if (tid < HH) {
      float v = xv[tid];
      o_delta[(size_t)t * HH + tid] = v;
      dprev[tid] = v;
      ss = v * v;
    }
    red[tid] = ss;
    __syncthreads();
    for (int s = 512; s > 0; s >>= 1) {
      if (tid < s) red[tid] += red[tid + s];
      __syncthreads();
    }
    if (tid == 0) o_dn[t] = sqrtf(red[0]);
    __syncthreads();
  }
}

// --------------------------------------------------------------- launcher
extern "C" void kernel_launch(void* const* d_in, const int* in_sizes, int n_in,
                              void* d_out, int out_size, void* d_ws,
                              size_t ws_size, hipStream_t stream) {
  const float* x    = (const float*)d_in[0];
  const float* dout = (const float*)d_in[1];
  const float* Wx   = (const float*)d_in[2];
  const float* Wh   = (const float*)d_in[3];
  const float* b    = (const float*)d_in[4];
  const float* lns  = (const float*)d_in[5];
  const float* lnb  = (const float*)d_in[6];
  const float* Wo   = (const float*)d_in[7];
  const float* bo   = (const float*)d_in[8];

  float* out      = (float*)d_out;
  float* o_output = out;                 // 512*128
  float* o_prevh  = out + 65536;         // 512*256
  float* o_delta  = out + 196608;        // 512*256
  float* o_inst   = out + 327680;        // 512*256
  float* o_hn     = out + 458752;        // 512
  float* o_dn     = out + 459264;        // 512
  float* o_pj     = out + 459776;        // 512
  float* o_jac    = out + 460288;        // 512*256*256

  float* ws     = (float*)d_ws;
  float* gx_all = ws;                    // 512*768
  float* cA     = ws + 393216;           // 512*256 each
  float* cB     = cA + 131072;
  float* cC     = cB + 131072;
  float* cZ     = cC + 131072;

  k_gx<<<(TT * H3 + 255) / 256, 256, 0, stream>>>(x, Wx, b, gx_all);
  k_recur<<<1, 768, 0, stream>>>(dout, Wh, lns, lnb, Wo, bo, gx_all, o_output,
                                 o_prevh, o_inst, o_hn, cA, cB, cC, cZ);
  k_jac<<<TT, 256, 0, stream>>>(Wh, cA, cB, cC, cZ, o_jac);
  size_t lds3 =
      (size_t)(256 * PSTR + 2 * 256 * BT + 1024) * sizeof(float);  // ~298KB
  k_prodjac<<<1, 1024, lds3, stream>>>(o_jac, o_pj);
  size_t lds4 = (size_t)(256 * PSTR + 256 + 256 + 1024) * sizeof(float);
  k_delta<<<1, 1024, lds4, stream>>>(o_jac, o_inst, o_delta, o_dn);
}